// HGB_final_51582557225172
// MI455X (gfx1250) — compile-verified
//
#include <hip/hip_runtime.h>
#include <hip/hip_bf16.h>

// HGB GAT forward for MI455X (gfx1250, wave32).
// f32 WMMA (V_WMMA_F32_16X16X4_F32) for the two 100000x128x64 GEMMs,
// LDS-staged weights, native f32 global atomics for edge aggregation.

typedef __attribute__((ext_vector_type(2))) float v2f;
typedef __attribute__((ext_vector_type(8))) float v8f;

#define D_IN 128
#define C_OUT 64
#define NEG_SLOPE 0.2f
#define L2_EPS 1e-12f

// ---------------------------------------------------------------------------
// memset helper (graph-safe zero init of workspace accumulators)
// ---------------------------------------------------------------------------
__global__ void hgb_zero_kernel(float* __restrict__ p, long long n) {
    long long i = (long long)blockIdx.x * blockDim.x + threadIdx.x;
    long long stride = (long long)gridDim.x * blockDim.x;
    for (; i < n; i += stride) p[i] = 0.0f;
}

// ---------------------------------------------------------------------------
// Tiny parameter prep:
//   w_l[k] = sum_n W[n][k]*a_l[n]   (so h_l = head_feature . w_l)
//   w_r[k] = sum_n W[n][k]*a_r[n]
//   het[t] = edge_emb[t] . (W_e @ a_e)
// one block of 128 threads
// ---------------------------------------------------------------------------
__global__ void hgb_prep_kernel(const float* __restrict__ W,
                                const float* __restrict__ W_e,
                                const float* __restrict__ edge_emb,
                                const float* __restrict__ a_l,
                                const float* __restrict__ a_r,
                                const float* __restrict__ a_e,
                                float* __restrict__ w_l,
                                float* __restrict__ w_r,
                                float* __restrict__ het) {
    __shared__ float v[C_OUT];
    const int t = threadIdx.x;
    if (t < D_IN) {
        float sl = 0.f, sr = 0.f;
        for (int n = 0; n < C_OUT; ++n) {
            float w = W[n * D_IN + t];
            sl += w * a_l[n];
            sr += w * a_r[n];
        }
        w_l[t] = sl;
        w_r[t] = sr;
    }
    if (t < 64) {
        float s = 0.f;
        for (int j = 0; j < 64; ++j) s += W_e[t * 64 + j] * a_e[j];
        v[t] = s;
    }
    __syncthreads();
    if (t < 8) {
        float s = 0.f;
        for (int i = 0; i < 64; ++i) s += edge_emb[t * 64 + i] * v[i];
        het[t] = s;
    }
}

// ---------------------------------------------------------------------------
// Y[M,64] = X[M,128] @ Wt[64,128]^T + bias[64]
// block = 256 threads = 8 wave32s; each wave owns a 16x64 output tile;
// Wt (32KB) staged in LDS; K loop of 32 steps of V_WMMA_F32_16X16X4_F32.
// f32 WMMA layouts (wave32):
//   A 16x4:  vgpr j, lanes0-15 -> A[m=lane][k=K0+j]; lanes16-31 -> k=K0+2+j
//   B 4x16:  vgpr j, lanes0-15 -> B[k=K0+j][n=lane]; lanes16-31 -> k=K0+2+j
//   D 16x16: vgpr j, lanes0-15 -> D[m=j][n=lane];    lanes16-31 -> m=8+j
// ---------------------------------------------------------------------------
__device__ __forceinline__ v8f splat8(float x) {
    v8f r;
#pragma unroll
    for (int i = 0; i < 8; ++i) r[i] = x;
    return r;
}

__global__ void hgb_gemm_xwt_kernel(const float* __restrict__ X,
                                    const float* __restrict__ Wt,
                                    const float* __restrict__ bias,
                                    float* __restrict__ Y, int M) {
    __shared__ float sW[C_OUT * D_IN];   // 32 KB
    const int tid = threadIdx.x;

    // cooperative LDS fill of Wt (row-major [64][128])
    {
        const float4* src = (const float4*)Wt;
        float4* dst = (float4*)sW;
#pragma unroll
        for (int i = 0; i < (C_OUT * D_IN / 4) / 256; ++i)
            dst[tid + i * 256] = src[tid + i * 256];
    }
    __syncthreads();

    const int lane = tid & 31;
    const int wave = tid >> 5;
    const int grp = lane >> 4;     // 0: lanes 0-15, 1: lanes 16-31
    const int half = lane & 15;
    const int rowBase = blockIdx.x * 128 + wave * 16;

    // clamp A-row so EXEC stays all-ones for WMMA (stores masked later)
    int arow = rowBase + half;
    if (arow >= M) arow = M - 1;
    const float* Xrow = X + (size_t)arow * D_IN;

    v8f acc0 = splat8(bias[0 + half]);
    v8f acc1 = splat8(bias[16 + half]);
    v8f acc2 = splat8(bias[32 + half]);
    v8f acc3 = splat8(bias[48 + half]);

#pragma unroll 4
    for (int kk = 0; kk < D_IN / 4; ++kk) {
        const int kb = kk * 4 + grp * 2;       // even -> 8B aligned
        v2f a = *(const v2f*)(Xrow + kb);
        v2f b0 = *(const v2f*)(sW + (0 + half) * D_IN + kb);
        acc0 = __builtin_amdgcn_wmma_f32_16x16x4_f32(false, a, false, b0,
                                                     (short)0, acc0, false, false);
        v2f b1 = *(const v2f*)(sW + (16 + half) * D_IN + kb);
        acc1 = __builtin_amdgcn_wmma_f32_16x16x4_f32(false, a, false, b1,
                                                     (short)0, acc1, false, false);
        v2f b2 = *(const v2f*)(sW + (32 + half) * D_IN + kb);
        acc2 = __builtin_amdgcn_wmma_f32_16x16x4_f32(false, a, false, b2,
                                                     (short)0, acc2, false, false);
        v2f b3 = *(const v2f*)(sW + (48 + half) * D_IN + kb);
        acc3 = __builtin_amdgcn_wmma_f32_16x16x4_f32(false, a, false, b3,
                                                     (short)0, acc3, false, false);
    }

#pragma unroll
    for (int j = 0; j < 8; ++j) {
        const int row = rowBase + grp * 8 + j;
        if (row < M) {
            float* yr = Y + (size_t)row * C_OUT;
            yr[0 + half]  = acc0[j];
            yr[16 + half] = acc1[j];
            yr[32 + half] = acc2[j];
            yr[48 + half] = acc3[j];
        }
    }
}

// ---------------------------------------------------------------------------
// out[i] = X[i,:] . w   (one wave per node, lanes cover 128 with 4 each)
// ---------------------------------------------------------------------------
__global__ void hgb_rowdot_kernel(const float* __restrict__ X,
                                  const float* __restrict__ w,
                                  float* __restrict__ out, int M) {
    const int lane = threadIdx.x & 31;
    const float w0 = w[lane], w1 = w[lane + 32], w2 = w[lane + 64], w3 = w[lane + 96];
    int wid = blockIdx.x * (blockDim.x >> 5) + (threadIdx.x >> 5);
    const int nw = gridDim.x * (blockDim.x >> 5);
    for (int i = wid; i < M; i += nw) {
        const float* row = X + (size_t)i * D_IN;
        float s = row[lane] * w0 + row[lane + 32] * w1 +
                  row[lane + 64] * w2 + row[lane + 96] * w3;
#pragma unroll
        for (int o = 16; o > 0; o >>= 1) s += __shfl_down(s, o, 32);
        if (lane == 0) out[i] = s;
    }
}

// ordered-uint encoding of float for monotone atomicMax
__device__ __forceinline__ unsigned f2ord(float f) {
    unsigned u = __float_as_uint(f);
    return (u & 0x80000000u) ? ~u : (u | 0x80000000u);
}
__device__ __forceinline__ float ord2f(unsigned u) {
    return (u & 0x80000000u) ? __uint_as_float(u ^ 0x80000000u)
                             : __uint_as_float(~u);
}

// ---------------------------------------------------------------------------
// edge pass A: att logit + per-head-node segment max (thread per edge)
// ---------------------------------------------------------------------------
__global__ void hgb_edge_max_kernel(const int* __restrict__ head,
                                    const int* __restrict__ tail,
                                    const int* __restrict__ etype,
                                    const float* __restrict__ h_l,
                                    const float* __restrict__ h_r,
                                    const float* __restrict__ het,
                                    unsigned* __restrict__ m_ord, int E_) {
    int i = blockIdx.x * blockDim.x + threadIdx.x;
    const int stride = gridDim.x * blockDim.x;
    for (; i < E_; i += stride) {
        const int h = head[i];
        float a = h_l[h] + h_r[tail[i]] + het[etype[i]];
        a = (a >= 0.f) ? a : NEG_SLOPE * a;
        atomicMax(&m_ord[h], f2ord(a));
    }
}

// ---------------------------------------------------------------------------
// edge pass B: ex = exp(att - m[head]); denom[head] += ex;
//              msg[head,:] += ex * h_tail[tail,:]     (wave per edge)
// ---------------------------------------------------------------------------
__global__ void hgb_edge_acc_kernel(const int* __restrict__ head,
                                    const int* __restrict__ tail,
                                    const int* __restrict__ etype,
                                    const float* __restrict__ h_l,
                                    const float* __restrict__ h_r,
                                    const float* __restrict__ het,
                                    const unsigned* __restrict__ m_ord,
                                    const float* __restrict__ h_tail,
                                    float* __restrict__ msg,
                                    float* __restrict__ denom, int E_) {
    const int lane = threadIdx.x & 31;
    int wid = blockIdx.x * (blockDim.x >> 5) + (threadIdx.x >> 5);
    const int nw = gridDim.x * (blockDim.x >> 5);
    for (int i = wid; i < E_; i += nw) {
        const int h = head[i];
        const int t = tail[i];
        float a = h_l[h] + h_r[t] + het[etype[i]];
        a = (a >= 0.f) ? a : NEG_SLOPE * a;
        const float ex = __expf(a - ord2f(m_ord[h]));
        if (lane == 0) unsafeAtomicAdd(&denom[h], ex);
        const float* ht = h_tail + (size_t)t * C_OUT;
        float* mg = msg + (size_t)h * C_OUT;
        unsafeAtomicAdd(&mg[lane],      ex * ht[lane]);
        unsafeAtomicAdd(&mg[lane + 32], ex * ht[lane + 32]);
    }
}

// ---------------------------------------------------------------------------
// epilogue: out = softmax( l2norm( msg/denom + residual ) )   (wave per node)
// ---------------------------------------------------------------------------
__global__ void hgb_finalize_kernel(const float* __restrict__ msg,
                                    const float* __restrict__ denom,
                                    const float* __restrict__ res,
                                    float* __restrict__ out, int M) {
    const int lane = threadIdx.x & 31;
    int wid = blockIdx.x * (blockDim.x >> 5) + (threadIdx.x >> 5);
    const int nw = gridDim.x * (blockDim.x >> 5);
    for (int i = wid; i < M; i += nw) {
        const float d = denom[i];
        const float inv = (d > 0.f) ? (1.f / d) : 0.f;
        const size_t b = (size_t)i * C_OUT;
        float v0 = msg[b + lane] * inv + res[b + lane];
        float v1 = msg[b + lane + 32] * inv + res[b + lane + 32];
        // L2 norm of row
        float ss = v0 * v0 + v1 * v1;
#pragma unroll
        for (int o = 16; o > 0; o >>= 1) ss += __shfl_xor(ss, o, 32);
        const float innorm = 1.f / fmaxf(sqrtf(ss), L2_EPS);
        const float h0 = v0 * innorm, h1 = v1 * innorm;
        // softmax over 64 channels
        float mx = fmaxf(h0, h1);
#pragma unroll
        for (int o = 16; o > 0; o >>= 1) mx = fmaxf(mx, __shfl_xor(mx, o, 32));
        const float e0 = __expf(h0 - mx), e1 = __expf(h1 - mx);
        float s = e0 + e1;
#pragma unroll
        for (int o = 16; o > 0; o >>= 1) s += __shfl_xor(s, o, 32);
        const float rs = 1.f / s;
        out[b + lane] = e0 * rs;
        out[b + lane + 32] = e1 * rs;
    }
}

// ---------------------------------------------------------------------------
extern "C" void kernel_launch(void* const* d_in, const int* in_sizes, int n_in,
                              void* d_out, int out_size, void* d_ws, size_t ws_size,
                              hipStream_t stream) {
    const float* head_feature = (const float*)d_in[0];
    const float* tail_feature = (const float*)d_in[1];
    const float* W            = (const float*)d_in[2];
    const float* W_e          = (const float*)d_in[3];
    const float* edge_emb     = (const float*)d_in[4];
    const float* a_l          = (const float*)d_in[5];
    const float* a_r          = (const float*)d_in[6];
    const float* a_e          = (const float*)d_in[7];
    const float* res_W        = (const float*)d_in[8];
    const float* res_b        = (const float*)d_in[9];
    const int*   edge_index   = (const int*)d_in[10];
    const int*   tmp_edge     = (const int*)d_in[11];

    const int Nh = in_sizes[0] / D_IN;
    const int Nt = in_sizes[1] / D_IN;
    const int E_ = in_sizes[11];
    const int* head = edge_index;
    const int* tail = edge_index + E_;

    // workspace layout (floats) — total ~79 MB
    float* ws = (float*)d_ws;
    float* ws_h_tail = ws;                              // Nt*64
    float* ws_res    = ws_h_tail + (size_t)Nt * C_OUT;  // Nh*64
    float* ws_msg    = ws_res + (size_t)Nh * C_OUT;     // Nh*64  (zeroed)
    float* ws_denom  = ws_msg + (size_t)Nh * C_OUT;     // Nh     (zeroed)
    unsigned* ws_mord = (unsigned*)(ws_denom + Nh);     // Nh     (zeroed)
    float* ws_h_l    = (float*)(ws_mord + Nh);          // Nh
    float* ws_h_r    = ws_h_l + Nh;                     // Nt
    float* ws_w_l    = ws_h_r + Nt;                     // 128
    float* ws_w_r    = ws_w_l + 128;                    // 128
    float* ws_het    = ws_w_r + 128;                    // 64 (8 used)
    float* ws_zbias  = ws_het + 64;                     // 64 zeros (zeroed)

    // zero msg/denom/m_ord/.../zbias in one contiguous sweep
    const long long nzero = (long long)(ws_zbias + 64 - ws_msg);
    hgb_zero_kernel<<<2048, 256, 0, stream>>>(ws_msg, nzero);

    // tiny parameter prep
    hgb_prep_kernel<<<1, 128, 0, stream>>>(W, W_e, edge_emb, a_l, a_r, a_e,
                                           ws_w_l, ws_w_r, ws_het);

    // WMMA GEMMs: h_tail = tail_feature @ W^T ; res = head_feature @ res_W^T + res_b
    hgb_gemm_xwt_kernel<<<(Nt + 127) / 128, 256, 0, stream>>>(
        tail_feature, W, ws_zbias, ws_h_tail, Nt);
    hgb_gemm_xwt_kernel<<<(Nh + 127) / 128, 256, 0, stream>>>(
        head_feature, res_W, res_b, ws_res, Nh);

    // per-node attention scalars (matvec form)
    hgb_rowdot_kernel<<<512, 256, 0, stream>>>(head_feature, ws_w_l, ws_h_l, Nh);
    hgb_rowdot_kernel<<<512, 256, 0, stream>>>(tail_feature, ws_w_r, ws_h_r, Nt);

    // edge passes: segment max, then exp-weighted accumulation
    hgb_edge_max_kernel<<<2048, 256, 0, stream>>>(head, tail, tmp_edge,
                                                  ws_h_l, ws_h_r, ws_het,
                                                  ws_mord, E_);
    hgb_edge_acc_kernel<<<2048, 256, 0, stream>>>(head, tail, tmp_edge,
                                                  ws_h_l, ws_h_r, ws_het,
                                                  ws_mord, ws_h_tail,
                                                  ws_msg, ws_denom, E_);

    // epilogue: normalize + softmax
    hgb_finalize_kernel<<<1024, 256, 0, stream>>>(ws_msg, ws_denom, ws_res,
                                                  (float*)d_out, Nh);
}